// LaplacianRegularization_32744830665388
// MI455X (gfx1250) — compile-verified
//
#include <hip/hip_runtime.h>
#include <hip/hip_bf16.h>

// Dirichlet energy: mean_e w_e * ||h[src_e] - h[dst_e]||^2
// h: [100000,128] f32, edge_index: [2,E] i64, edge_weight: [E] f32 -> scalar f32
//
// Design (MI455X): pure gather-bandwidth problem (0.23 GFLOP vs ~630MB of row
// gathers; h fits in 192MB L2). One wave32 per edge so each row read is one
// fully-coalesced 512B global_load_b128 sweep. Per-lane accumulators across
// edges; single WMMA-based (v_wmma_f32_16x16x4_f32) block reduction at the end.

typedef __attribute__((ext_vector_type(2))) float v2f;
typedef __attribute__((ext_vector_type(8))) float v8f;

#define NBLOCKS 1024
#define BLOCK   256   // 8 waves (wave32)

// Reduce BLOCK per-thread floats to one float using the matrix pipe.
// A = ones (16x4), B = 4x16 slabs of the 256 values from LDS, C accumulates
// 4 chunks -> each lane n holds colsum over 16 values; 4 xor-shuffles finish.
// Result valid in lane 0 of wave 0.
__device__ __forceinline__ float block_reduce_wmma(float acc, float* lds) {
    const int tid = threadIdx.x;
    lds[tid] = acc;
    __syncthreads();
    float total = 0.0f;
    if (tid < 32) {                 // whole wave 0, EXEC all ones
        v2f a; a[0] = 1.0f; a[1] = 1.0f;   // A[m,k] = 1 for all m,k
        v8f c = {};
#pragma unroll
        for (int chunk = 0; chunk < 4; ++chunk) {
            v2f b;
            // B 4x16 f32 layout: v0 = {K0 lanes0-15, K1 lanes16-31},
            //                    v1 = {K2 lanes0-15, K3 lanes16-31}
            b[0] = lds[chunk * 64 + tid];
            b[1] = lds[chunk * 64 + 32 + tid];
            // D[m,n] += sum_k B[k,n]  (column sums, replicated over rows)
            c = __builtin_amdgcn_wmma_f32_16x16x4_f32(
                    false, a, false, b, (short)0, c, false, false);
        }
        float s = c[0];             // colsum for column (lane & 15)
#pragma unroll
        for (int m = 1; m <= 8; m <<= 1)
            s += __shfl_xor(s, m, 32);   // sum the 16 column sums
        total = s;
    }
    return total;
}

__global__ void __launch_bounds__(BLOCK)
dirichlet_partial(const float* __restrict__ h,
                  const long long* __restrict__ ei,   // [2,E] row-major
                  const float* __restrict__ w,
                  float* __restrict__ partials,
                  int E)
{
    __shared__ float lds[BLOCK];
    const int lane   = threadIdx.x & 31;
    const int gwave  = (blockIdx.x * BLOCK + threadIdx.x) >> 5;
    const int nwaves = (gridDim.x * BLOCK) >> 5;

    float acc = 0.0f;
    for (int e = gwave; e < E; e += nwaves) {
        // wave-uniform scalars (hardware-broadcast loads)
        const long long s = ei[e];
        const long long d = ei[E + e];
        const float     we = w[e];

        // 32 lanes x float4 = one 512B row each, fully coalesced b128 loads
        const float4* ps = (const float4*)(h + s * 128) + lane;
        const float4* pd = (const float4*)(h + d * 128) + lane;
        const float4 va = *ps;
        const float4 vb = *pd;

        const float d0 = va.x - vb.x;
        const float d1 = va.y - vb.y;
        const float d2 = va.z - vb.z;
        const float d3 = va.w - vb.w;
        float t = d0 * d0;
        t = fmaf(d1, d1, t);
        t = fmaf(d2, d2, t);
        t = fmaf(d3, d3, t);
        acc = fmaf(we, t, acc);     // w_e distributes over the dim sum
    }

    const float total = block_reduce_wmma(acc, lds);
    if (threadIdx.x == 0) partials[blockIdx.x] = total;
}

__global__ void __launch_bounds__(BLOCK)
dirichlet_final(const float* __restrict__ partials,
                float* __restrict__ out,
                int n, float scale)
{
    __shared__ float lds[BLOCK];
    float acc = 0.0f;
    for (int i = threadIdx.x; i < n; i += BLOCK) acc += partials[i];
    const float total = block_reduce_wmma(acc, lds);
    if (threadIdx.x == 0) out[0] = total * scale;   // mean = sum / E
}

extern "C" void kernel_launch(void* const* d_in, const int* in_sizes, int n_in,
                              void* d_out, int out_size, void* d_ws, size_t ws_size,
                              hipStream_t stream) {
    const float*     h  = (const float*)d_in[0];
    const long long* ei = (const long long*)d_in[1];   // int64 per reference
    const float*     w  = (const float*)d_in[2];
    const int        E  = in_sizes[2];                 // edge_weight count

    float* partials = (float*)d_ws;                    // NBLOCKS * 4 bytes

    dirichlet_partial<<<NBLOCKS, BLOCK, 0, stream>>>(h, ei, w, partials, E);
    dirichlet_final<<<1, BLOCK, 0, stream>>>(partials, (float*)d_out,
                                             NBLOCKS, 1.0f / (float)E);
}